// BLinear_15625091023046
// MI455X (gfx1250) — compile-verified
//
#include <hip/hip_runtime.h>
#include <math.h>

#define NBS  2048   // NB*NS
#define NIN  256
#define NOUT 256

typedef __attribute__((ext_vector_type(2))) float v2f;
typedef __attribute__((ext_vector_type(8))) float v8f;

// ---------------------------------------------------------------------------
// Kernel 1: ew[o,i] = exp(w_lsigma[o,i])  (64K elements -> workspace)
// ---------------------------------------------------------------------------
__global__ void ew_exp_kernel(const float* __restrict__ wls,
                              float* __restrict__ ew) {
    int i = blockIdx.x * blockDim.x + threadIdx.x;
    if (i < NOUT * NIN) ew[i] = expf(wls[i]);
}

// ---------------------------------------------------------------------------
// Kernel 2: out[bs,o] = (x @ w_mu^T)[bs,o] + b_mu[o] + exp(b_lsigma[o])*r2[bs,o]
// One wave per 16x16 output tile, f32 WMMA (16x16x4), K-loop of 64.
// A (16x4 f32): lane = (k>=2)*16 + m, vgpr j holds k = 2*(lane/16)+j.
// B (4x16 f32): lane = (k>=2)*16 + n, vgpr j holds k = 2*(lane/16)+j; B[k,n]=w_mu[n,k].
// C/D (16x16 f32): vgpr v, lanes 0-15: (M=v, N=lane); lanes 16-31: (M=v+8, N=lane-16).
// ---------------------------------------------------------------------------
__global__ void mean_wmma_kernel(const float* __restrict__ x,
                                 const float* __restrict__ wmu,
                                 const float* __restrict__ bmu,
                                 const float* __restrict__ blsig,
                                 const float* __restrict__ r2,
                                 float* __restrict__ out) {
    const int lane = threadIdx.x & 31;
    const int wave = threadIdx.x >> 5;
    const int tile = blockIdx.x * 8 + wave;   // 0..2047
    const int tile_m = tile >> 4;             // 0..127 (bs tiles)
    const int tile_n = tile & 15;             // 0..15  (o tiles)
    const int bs0 = tile_m * 16;
    const int o0  = tile_n * 16;
    const int half = lane >> 4;               // 0 or 1
    const int l16  = lane & 15;               // m for A, n for B
    const int kk   = half * 2;

    v8f c = {0.f, 0.f, 0.f, 0.f, 0.f, 0.f, 0.f, 0.f};

    const float* xrow = x   + (bs0 + l16) * NIN;  // A row (M = l16)
    const float* wrow = wmu + (o0  + l16) * NIN;  // B col (N = l16), w_mu[n,k]

#pragma unroll 4
    for (int k0 = 0; k0 < NIN; k0 += 4) {
        v2f a = *(const v2f*)(xrow + k0 + kk);   // 8B-aligned (k0+kk even)
        v2f b = *(const v2f*)(wrow + k0 + kk);
        c = __builtin_amdgcn_wmma_f32_16x16x4_f32(
                /*neg_a=*/false, a, /*neg_b=*/false, b,
                /*c_mod=*/(short)0, c, /*reuse_a=*/false, /*reuse_b=*/false);
    }

    // Epilogue: fuse deterministic bias + reparameterized bias noise.
    const int   col     = o0 + l16;
    const float bias_mu = bmu[col];
    const float bsig    = expf(blsig[col]);
#pragma unroll
    for (int v = 0; v < 8; ++v) {
        const int row = bs0 + v + half * 8;
        const float bias = bias_mu + bsig * r2[row * NOUT + col];
        out[row * NOUT + col] = c[v] + bias;
    }
}

// ---------------------------------------------------------------------------
// Kernel 3: out[bs,o] += sum_i r1[bs,o,i] * ew[o,i] * x[bs,i]
// One wave per 4 consecutive rows (same bs -> x loaded once).
// Per row each lane does two float4 (B128) loads of r1 at i=4*lane and
// i=128+4*lane -> 512B contiguous per load instruction across the wave.
// ---------------------------------------------------------------------------
__global__ void noise_kernel(const float* __restrict__ r1,
                             const float* __restrict__ ew,
                             const float* __restrict__ x,
                             float* __restrict__ out) {
    const int lane = threadIdx.x & 31;
    const int wid  = (blockIdx.x * blockDim.x + threadIdx.x) >> 5;
    const int row0 = wid * 4;                 // rows row0..row0+3, same bs
    const int bs   = row0 >> 8;

    const float4* xv = (const float4*)(x + bs * NIN);
    const float4  xa = xv[lane];              // i = 4*lane .. 4*lane+3
    const float4  xb = xv[lane + 32];         // i = 128+4*lane ..

#pragma unroll
    for (int rr = 0; rr < 4; ++rr) {
        const int row = row0 + rr;
        const int o   = row & 255;
        const float4* rv = (const float4*)(r1 + (size_t)row * NIN);
        const float4* ev = (const float4*)(ew + o * NIN);
        const float4 ra = rv[lane];
        const float4 rb = rv[lane + 32];
        const float4 ea = ev[lane];
        const float4 eb = ev[lane + 32];

        float acc = (ra.x * ea.x) * xa.x;
        acc = fmaf(ra.y * ea.y, xa.y, acc);
        acc = fmaf(ra.z * ea.z, xa.z, acc);
        acc = fmaf(ra.w * ea.w, xa.w, acc);
        acc = fmaf(rb.x * eb.x, xb.x, acc);
        acc = fmaf(rb.y * eb.y, xb.y, acc);
        acc = fmaf(rb.z * eb.z, xb.z, acc);
        acc = fmaf(rb.w * eb.w, xb.w, acc);

        // wave32 all-reduce
#pragma unroll
        for (int off = 16; off > 0; off >>= 1)
            acc += __shfl_xor(acc, off, 32);

        if (lane == 0) out[row] += acc;
    }
}

// ---------------------------------------------------------------------------
extern "C" void kernel_launch(void* const* d_in, const int* in_sizes, int n_in,
                              void* d_out, int out_size, void* d_ws, size_t ws_size,
                              hipStream_t stream) {
    (void)in_sizes; (void)n_in; (void)out_size; (void)ws_size;

    const float* x     = (const float*)d_in[0];
    const float* w_mu  = (const float*)d_in[1];
    const float* w_ls  = (const float*)d_in[2];
    const float* b_mu  = (const float*)d_in[3];
    const float* b_ls  = (const float*)d_in[4];
    const float* r1    = (const float*)d_in[5];
    const float* r2    = (const float*)d_in[6];
    float*       out   = (float*)d_out;
    float*       ew    = (float*)d_ws;        // 256 KB exp(w_lsigma) table

    // 1) exp table (must finish before kernel 3; stream order guarantees it)
    ew_exp_kernel<<<(NOUT * NIN + 255) / 256, 256, 0, stream>>>(w_ls, ew);

    // 2) WMMA mean GEMM + bias, full overwrite of d_out (2048 tiles, 8 waves/block)
    mean_wmma_kernel<<<2048 / 8, 256, 0, stream>>>(x, w_mu, b_mu, b_ls, r2, out);

    // 3) streaming noise contraction: 524288 rows / 4 rows-per-wave / 8 waves-per-block
    noise_kernel<<<(NBS * NOUT / 4) / 8, 256, 0, stream>>>(r1, ew, x, out);
}